// Converse2D_3015067042146
// MI455X (gfx1250) — compile-verified
//
#include <hip/hip_runtime.h>
#include <math.h>

// ---------------------------------------------------------------------------
// Converse2D on gfx1250: spectral closed form reduced to
//   out = ifft2_272( tile2x2(fft2_136(wrap_pad(x))) * G[c] ) cropped,
// with all DFTs executed as f32 GEMMs on V_WMMA_F32_16X16X4_F32.
// Round 4: force tile-validity guards into SGPRs (readfirstlane) so WMMAs are
// wrapped in scalar branches, never EXEC predication (ISA: EXEC must be all
// ones for WMMA); compile-time GEMM dims; fully unrolled accumulator stores.
// ---------------------------------------------------------------------------

typedef __attribute__((ext_vector_type(2))) float f32x2;
typedef __attribute__((ext_vector_type(8))) float f32x8;

#define IMG   128
#define HP    136      // 128 + 2*PAD
#define HPAD  144      // 136 padded to 9*16 for WMMA tiling
#define S2    272      // 2*HP, = 17*16 exactly
#define OUTW  256
#define CH    64
#define NPLANES 256    // N*C = 4*64
#define PSZ144 (HPAD*HPAD)   // 20736
#define PSZ272 (S2*S2)       // 73984
#define TWO_PI 6.28318530717958647692f

__device__ inline bool uni(bool b) {           // force condition into an SGPR
  return __builtin_amdgcn_readfirstlane((int)b) != 0;
}

// ---------------------------- table builders -------------------------------

__global__ void build_W144(float* Wr, float* Wi) {
  int idx = blockIdx.x * blockDim.x + threadIdx.x;
  if (idx >= PSZ144) return;
  int u = idx / HPAD, k = idx - u * HPAD;
  float cr = 0.f, ci = 0.f;
  if (u < HP && k < HP) {
    int m = (u * k) % HP;                       // exact reduced phase index
    float ang = TWO_PI * (float)m / (float)HP;  // forward DFT: e^{-i ang}
    float s, c; sincosf(ang, &s, &c);
    cr = c; ci = -s;
  }
  Wr[idx] = cr; Wi[idx] = ci;
}

__global__ void build_V272(float* Vr, float* Vi) {
  int idx = blockIdx.x * blockDim.x + threadIdx.x;
  if (idx >= PSZ272) return;
  int u = idx / S2, k = idx - u * S2;
  int m = (u * k) % S2;
  float ang = TWO_PI * (float)m / (float)S2;    // inverse DFT: e^{+i ang}
  float s, c; sincosf(ang, &s, &c);
  Vr[idx] = c; Vi[idx] = s;
}

// FB[c,u,v] = sum_{kh,kw} w[c,kh,kw] * e^{-2pi i (u(kh-2)+v(kw-2))/272}
// (the roll(-K//2) in _p2o places tap (kh,kw) at (kh-2, kw-2) mod 272)
__global__ void build_FB(const float* __restrict__ w, float* __restrict__ G) {
  int idx = blockIdx.x * blockDim.x + threadIdx.x;
  if (idx >= CH * PSZ272) return;
  int c = idx / PSZ272, rem = idx - c * PSZ272;
  int u = rem / S2, v = rem - u * S2;
  float fr = 0.f, fi = 0.f;
  for (int kh = 0; kh < 5; ++kh)
    for (int kw = 0; kw < 5; ++kw) {
      int t = (u * (kh - 2) + v * (kw - 2)) % S2;
      if (t < 0) t += S2;
      float ang = TWO_PI * (float)t / (float)S2;
      float s, cc; sincosf(ang, &s, &cc);
      float wv = w[c * 25 + kh * 5 + kw];
      fr += wv * cc;
      fi -= wv * s;
    }
  G[(size_t)idx * 2 + 0] = fr;
  G[(size_t)idx * 2 + 1] = fi;
}

__device__ inline void khat_axis(int uu, float& kr, float& ki) {
  // 1 + e^{-2pi i uu / 272}
  float ang = TWO_PI * (float)uu / (float)S2;
  float s, c; sincosf(ang, &s, &c);
  kr = 1.f + c; ki = -s;
}

// invW = mean_4 |FB|^2 ; M = mean_4 (FB * Khat), over the 2x2 aliases.
__global__ void alias_stats(const float* __restrict__ G, float* __restrict__ st) {
  int idx = blockIdx.x * blockDim.x + threadIdx.x;
  if (idx >= CH * HP * HP) return;
  int c = idx / (HP * HP), rem = idx - c * (HP * HP);
  int u = rem / HP, v = rem - u * HP;
  float invW = 0.f, Mr = 0.f, Mi = 0.f;
  for (int si = 0; si < 2; ++si)
    for (int sj = 0; sj < 2; ++sj) {
      int uu = u + si * HP, vv = v + sj * HP;
      size_t off = ((size_t)c * PSZ272 + (size_t)uu * S2 + vv) * 2;
      float br = G[off], bi = G[off + 1];
      invW += br * br + bi * bi;
      float kur, kui, kvr, kvi;
      khat_axis(uu, kur, kui);
      khat_axis(vv, kvr, kvi);
      float kr = kur * kvr - kui * kvi;
      float ki = kur * kvi + kui * kvr;
      Mr += br * kr - bi * ki;
      Mi += br * ki + bi * kr;
    }
  size_t so = (size_t)idx * 3;
  st[so] = 0.25f * invW; st[so + 1] = 0.25f * Mr; st[so + 2] = 0.25f * Mi;
}

// G = conj(FB)*(1-M)/(invW+beta) + Khat   (in place over the FB buffer)
__global__ void make_G(float* __restrict__ G, const float* __restrict__ st,
                       const float* __restrict__ bias) {
  int idx = blockIdx.x * blockDim.x + threadIdx.x;
  if (idx >= CH * PSZ272) return;
  int c = idx / PSZ272, rem = idx - c * PSZ272;
  int u = rem / S2, v = rem - u * S2;
  int fu = (u >= HP) ? u - HP : u;
  int fv = (v >= HP) ? v - HP : v;
  float beta = 1.f / (1.f + expf(9.f - bias[c])) + 1e-5f;
  size_t off = (size_t)idx * 2;
  float fbr = G[off], fbi = G[off + 1];
  size_t so = ((size_t)c * (HP * HP) + (size_t)fu * HP + fv) * 3;
  float invW = st[so], Mr = st[so + 1], Mi = st[so + 2];
  float d = 1.f / (invW + beta);
  float pr = (1.f - Mr) * d, pi = -Mi * d;
  // conj(FB)*(pr+i pi)
  float gr = fbr * pr + fbi * pi;
  float gi = fbr * pi - fbi * pr;
  float kur, kui, kvr, kvi;
  khat_axis(u, kur, kui);
  khat_axis(v, kvr, kvi);
  gr += kur * kvr - kui * kvi;
  gi += kur * kvi + kui * kvr;
  G[off] = gr; G[off + 1] = gi;
}

// ---------------------------- per-plane stages -----------------------------

// wrap-pad (PAD=4) into a zero-padded 144x144 tile
__global__ void pad_wrap(const float* __restrict__ x, float* __restrict__ Xp,
                         int base, int pcount) {
  int idx = blockIdx.x * blockDim.x + threadIdx.x;
  if (idx >= pcount * PSZ144) return;
  int p = idx / PSZ144, rem = idx - p * PSZ144;
  int i = rem / HPAD, j = rem - i * HPAD;
  float v = 0.f;
  if (i < HP && j < HP) {
    int pg = base + p;
    int si = (i + IMG - 4) & (IMG - 1);
    int sj = (j + IMG - 4) & (IMG - 1);
    v = x[(size_t)pg * (IMG * IMG) + (size_t)si * IMG + sj];
  }
  Xp[(size_t)p * PSZ144 + rem] = v;
}

// S = tile2x2(F) * G[c]
__global__ void make_S(const float* __restrict__ Fr, const float* __restrict__ Fi,
                       const float* __restrict__ G,
                       float* __restrict__ Sr, float* __restrict__ Si,
                       int base, int pcount) {
  int idx = blockIdx.x * blockDim.x + threadIdx.x;
  if (idx >= pcount * PSZ272) return;
  int p = idx / PSZ272, rem = idx - p * PSZ272;
  int u = rem / S2, v = rem - u * S2;
  int c = (base + p) % CH;
  int fu = (u >= HP) ? u - HP : u;
  int fv = (v >= HP) ? v - HP : v;
  size_t foff = (size_t)p * PSZ144 + (size_t)fu * HPAD + fv;
  float fr = Fr[foff], fi = Fi[foff];
  size_t goff = ((size_t)c * PSZ272 + rem) * 2;
  float gr = G[goff], gi = G[goff + 1];
  Sr[(size_t)p * PSZ272 + rem] = fr * gr - fi * gi;
  Si[(size_t)p * PSZ272 + rem] = fr * gi + fi * gr;
}

// ------------------------- WMMA complex GEMM -------------------------------
// One wave32 computes a 32x32 output block (2x2 tiles of 16x16) of
// C = A * B (complex, f32), stepping K by 4 with V_WMMA_F32_16X16X4_F32.
// Fragment layouts per ISA 7.12.2:
//  A 16x4 : lane%16 = M ; VGPR0 = {K0 | K2}, VGPR1 = {K1 | K3} split at lane16
//  B 4x16 : lane%16 = N ; same K striping
//  C      : 8 VGPRs, VGPR g -> M = g (lanes 0-15) / g+8 (lanes 16-31)
// The -Ai*Bi term uses a register negate (f32 WMMA NEG[0:1] must be 0).
// Edge-tile validity (9 / 17 tiles odd) goes through readfirstlane so it is
// provably scalar: WMMAs sit under s_cbranch, never under EXEC predication.

template <bool HAS_BI, bool WANT_CI>
__device__ inline void cplx_step(f32x8& cr, f32x8& ci,
                                 f32x2 ar_, f32x2 ai_, f32x2 nai_,
                                 f32x2 br_, f32x2 bi_) {
  cr = __builtin_amdgcn_wmma_f32_16x16x4_f32(false, ar_, false, br_,
                                             (short)0, cr, false, false);
  if constexpr (HAS_BI)
    cr = __builtin_amdgcn_wmma_f32_16x16x4_f32(false, nai_, false, bi_,
                                               (short)0, cr, false, false);
  if constexpr (WANT_CI) {
    ci = __builtin_amdgcn_wmma_f32_16x16x4_f32(false, ai_, false, br_,
                                               (short)0, ci, false, false);
    if constexpr (HAS_BI)
      ci = __builtin_amdgcn_wmma_f32_16x16x4_f32(false, ar_, false, bi_,
                                                 (short)0, ci, false, false);
  }
}

template <int MSZ, int NSZ, int KSZ, int LD, bool HAS_BI, bool WANT_CI, bool CROP>
__global__ void __launch_bounds__(256)
gemm_cplx(const float* __restrict__ Ar, const float* __restrict__ Ai, long aStride,
          const float* __restrict__ Br, const float* __restrict__ Bi, long bStride,
          float* Cr, float* Ci, long cStride,
          float* outBase, int chunkBase, float scale) {
  constexpr int ntN = (NSZ + 15) / 16;
  constexpr int ntM = (MSZ + 15) / 16;
  constexpr int npN = (ntN + 1) / 2;

  const int lane = threadIdx.x;                 // 0..31
  const int tn2  = blockIdx.x * 8 + threadIdx.y;  // 32-wide N block
  const int tm2  = blockIdx.y;                  // 32-wide M block
  const int p    = blockIdx.z;
  if (uni(tn2 >= npN)) return;                  // scalar branch

  const bool nok = uni(tn2 * 2 + 1 < ntN);      // second N tile valid (SGPR)
  const bool mok = uni(tm2 * 2 + 1 < ntM);      // second M tile valid (SGPR)

  const float* ar = Ar + (size_t)p * aStride;
  const float* ai = Ai + (size_t)p * aStride;
  const float* br = Br + (size_t)p * bStride;
  const float* bi = HAS_BI ? (Bi + (size_t)p * bStride) : nullptr;

  const int r0   = tm2 * 32 + (lane & 15);
  const int c0   = tn2 * 32 + (lane & 15);
  const int ksub = (lane >> 4) << 1;            // 0 or 2

  f32x8 cr00 = {}, cr01 = {}, cr10 = {}, cr11 = {};
  f32x8 ci00 = {}, ci01 = {}, ci10 = {}, ci11 = {};

  for (int k0 = 0; k0 < KSZ; k0 += 4) {
    const int kk = k0 + ksub;
    // A fragments: rows r0 and r0+16, reused across both N tiles
    f32x2 a0r, a0i, a1r = {}, a1i = {};
    a0r.x = ar[(size_t)r0 * LD + kk];        a0r.y = ar[(size_t)r0 * LD + kk + 1];
    a0i.x = ai[(size_t)r0 * LD + kk];        a0i.y = ai[(size_t)r0 * LD + kk + 1];
    if (mok) {
      a1r.x = ar[(size_t)(r0+16) * LD + kk]; a1r.y = ar[(size_t)(r0+16) * LD + kk + 1];
      a1i.x = ai[(size_t)(r0+16) * LD + kk]; a1i.y = ai[(size_t)(r0+16) * LD + kk + 1];
    }
    // B fragments: cols c0 and c0+16, reused across both M tiles
    f32x2 b0r, b1r = {}, b0i = {}, b1i = {};
    b0r.x = br[(size_t)kk * LD + c0];        b0r.y = br[(size_t)(kk+1) * LD + c0];
    if (nok) {
      b1r.x = br[(size_t)kk * LD + c0+16];   b1r.y = br[(size_t)(kk+1) * LD + c0+16];
    }
    if constexpr (HAS_BI) {
      b0i.x = bi[(size_t)kk * LD + c0];      b0i.y = bi[(size_t)(kk+1) * LD + c0];
      if (nok) {
        b1i.x = bi[(size_t)kk * LD + c0+16]; b1i.y = bi[(size_t)(kk+1) * LD + c0+16];
      }
    }
    const f32x2 na0i = -a0i, na1i = -a1i;

    cplx_step<HAS_BI, WANT_CI>(cr00, ci00, a0r, a0i, na0i, b0r, b0i);
    if (nok)        cplx_step<HAS_BI, WANT_CI>(cr01, ci01, a0r, a0i, na0i, b1r, b1i);
    if (mok)        cplx_step<HAS_BI, WANT_CI>(cr10, ci10, a1r, a1i, na1i, b0r, b0i);
    if (mok && nok) cplx_step<HAS_BI, WANT_CI>(cr11, ci11, a1r, a1i, na1i, b1r, b1i);
  }

  // ---- store 2x2 tiles (explicitly unrolled; no accumulator muxing) ----
  const int colA  = tn2 * 32 + (lane & 15);
  const int rbase = tm2 * 32 + ((lane >> 4) << 3);
  float* outp = CROP ? (outBase + (size_t)(chunkBase + p) * (OUTW * OUTW)) : nullptr;
  float* crp  = CROP ? nullptr : (Cr + (size_t)p * cStride);
  float* cip  = (!CROP && WANT_CI) ? (Ci + (size_t)p * cStride) : nullptr;

  auto store_tile = [&](const f32x8& aR, const f32x8& aI, int rb, int col) {
    if constexpr (CROP) {
      // fused crop [8:264) + 1/272^2 scale, straight into d_out
#pragma unroll
      for (int g = 0; g < 8; ++g) {
        int r = rb + g;
        if (r >= 8 && r < 8 + OUTW && col >= 8 && col < 8 + OUTW)
          outp[(size_t)(r - 8) * OUTW + (col - 8)] = aR[g] * scale;
      }
    } else {
#pragma unroll
      for (int g = 0; g < 8; ++g) {
        int r = rb + g;
        crp[(size_t)r * LD + col] = aR[g];
        if constexpr (WANT_CI) cip[(size_t)r * LD + col] = aI[g];
      }
    }
  };

  store_tile(cr00, ci00, rbase,      colA);
  if (nok)        store_tile(cr01, ci01, rbase,      colA + 16);
  if (mok)        store_tile(cr10, ci10, rbase + 16, colA);
  if (mok && nok) store_tile(cr11, ci11, rbase + 16, colA + 16);
}

// ------------------------------- launcher ----------------------------------

static inline int ceil_div(int a, int b) { return (a + b - 1) / b; }

static inline dim3 gemm_grid(int Msz, int Nsz, int pc) {
  int ntM = ceil_div(Msz, 16), ntN = ceil_div(Nsz, 16);
  int npM = ceil_div(ntM, 2),  npN = ceil_div(ntN, 2);
  return dim3(ceil_div(npN, 8), npM, pc);
}

extern "C" void kernel_launch(void* const* d_in, const int* in_sizes, int n_in,
                              void* d_out, int out_size, void* d_ws, size_t ws_size,
                              hipStream_t stream) {
  const float* x      = (const float*)d_in[0];   // (4,64,128,128)
  const float* weight = (const float*)d_in[1];   // (1,64,5,5)
  const float* bias   = (const float*)d_in[2];   // (1,64,1,1)
  float* out = (float*)d_out;                    // (4,64,256,256)
  float* ws  = (float*)d_ws;

  // ---- workspace layout (floats) ----
  const size_t o_Wr = 0;
  const size_t o_Wi = o_Wr + PSZ144;
  const size_t o_Vr = o_Wi + PSZ144;
  const size_t o_Vi = o_Vr + PSZ272;
  const size_t o_G  = o_Vi + PSZ272;                       // CH*PSZ272*2 (interleaved)
  const size_t o_st = o_G + (size_t)CH * PSZ272 * 2;       // CH*HP*HP*3
  const size_t o_ck = o_st + (size_t)CH * HP * HP * 3;
  const size_t perPlane = 5 * (size_t)PSZ144 + 4 * (size_t)PSZ272; // Xp,Ar,Ai,Fr,Fi,Sr,Si,Tr,Ti

  size_t availF = ws_size / sizeof(float);
  int P = 1;
  if (availF > o_ck + perPlane) {
    size_t pmax = (availF - o_ck) / perPlane;
    P = (int)(pmax > 16 ? 16 : pmax);
    if (P < 1) P = 1;
  }

  float* Wr = ws + o_Wr; float* Wi = ws + o_Wi;
  float* Vr = ws + o_Vr; float* Vi = ws + o_Vi;
  float* G  = ws + o_G;  float* st = ws + o_st;
  float* Xp = ws + o_ck;
  float* Ar = Xp + (size_t)P * PSZ144;
  float* Ai = Ar + (size_t)P * PSZ144;
  float* Fr = Ai + (size_t)P * PSZ144;
  float* Fi = Fr + (size_t)P * PSZ144;
  float* Sr = Fi + (size_t)P * PSZ144;
  float* Si = Sr + (size_t)P * PSZ272;
  float* Tr = Si + (size_t)P * PSZ272;
  float* Ti = Tr + (size_t)P * PSZ272;

  const int TB = 256;
  // ---- one-time spectral tables (L2-resident, reused by all 256 planes) ----
  build_W144<<<ceil_div(PSZ144, TB), TB, 0, stream>>>(Wr, Wi);
  build_V272<<<ceil_div(PSZ272, TB), TB, 0, stream>>>(Vr, Vi);
  build_FB<<<ceil_div(CH * PSZ272, TB), TB, 0, stream>>>(weight, G);
  alias_stats<<<ceil_div(CH * HP * HP, TB), TB, 0, stream>>>(G, st);
  make_G<<<ceil_div(CH * PSZ272, TB), TB, 0, stream>>>(G, st, bias);

  const float invD = 1.0f / ((float)S2 * (float)S2);
  const dim3 wgemm(32, 8, 1);

  for (int base = 0; base < NPLANES; base += P) {
    int pc = (NPLANES - base < P) ? (NPLANES - base) : P;

    pad_wrap<<<ceil_div(pc * PSZ144, TB), TB, 0, stream>>>(x, Xp, base, pc);

    // forward rows: (Ar + i Ai) = W * Xp   (B real)
    gemm_cplx<HPAD, HPAD, HPAD, HPAD, false, true, false>
        <<<gemm_grid(HPAD, HPAD, pc), wgemm, 0, stream>>>(
        Wr, Wi, 0, Xp, nullptr, PSZ144, Ar, Ai, PSZ144, nullptr, 0, 0.f);
    // forward cols: F = A * W
    gemm_cplx<HPAD, HPAD, HPAD, HPAD, true, true, false>
        <<<gemm_grid(HPAD, HPAD, pc), wgemm, 0, stream>>>(
        Ar, Ai, PSZ144, Wr, Wi, 0, Fr, Fi, PSZ144, nullptr, 0, 0.f);

    make_S<<<ceil_div(pc * PSZ272, TB), TB, 0, stream>>>(Fr, Fi, G, Sr, Si, base, pc);

    // inverse rows: T = V * S
    gemm_cplx<S2, S2, S2, S2, true, true, false>
        <<<gemm_grid(S2, S2, pc), wgemm, 0, stream>>>(
        Vr, Vi, 0, Sr, Si, PSZ272, Tr, Ti, PSZ272, nullptr, 0, 0.f);
    // inverse cols, real part only, fused crop+scale into d_out
    gemm_cplx<S2, S2, S2, S2, true, false, true>
        <<<gemm_grid(S2, S2, pc), wgemm, 0, stream>>>(
        Tr, Ti, PSZ272, Vr, Vi, 0, nullptr, nullptr, 0, out, base, invD);
  }
}